// QuantumErrorCorrector_8358006358656
// MI455X (gfx1250) — compile-verified
//
#include <hip/hip_runtime.h>

// CDNA5 / gfx1250: wave32, fp32 WMMA 16x16x4.
typedef float v2f  __attribute__((ext_vector_type(2)));
typedef float v8f  __attribute__((ext_vector_type(8)));
typedef float v2fa __attribute__((ext_vector_type(2), aligned(4)));

#define CORR_S  0.1f
#define ERR_T2  1e-12f   // (1e-6)^2 : compare squared norm, sqrt is monotonic

// Row content for logical output index o:
//  0-2  : syndromes      (W_syn row o,  bias b_syn[o])
//  3-11 : corr_out[j][k] (o = 3+3j+k -> W_corr row o-3, bias b_corr[o-3])
//  12-14: identity passthrough of the raw block, bias 0
//  15   : zero row
__device__ __forceinline__ void out_row(int o,
    const float* __restrict__ Ws, const float* __restrict__ bs,
    const float* __restrict__ Wc, const float* __restrict__ bc,
    float& w0, float& w1, float& w2, float& bias)
{
    w0 = w1 = w2 = bias = 0.0f;
    if (o < 3)        { w0 = Ws[o*3]; w1 = Ws[o*3+1]; w2 = Ws[o*3+2]; bias = bs[o]; }
    else if (o < 12)  { int q = o-3; w0 = Wc[q*3]; w1 = Wc[q*3+1]; w2 = Wc[q*3+2]; bias = bc[q]; }
    else if (o == 12) { w0 = 1.0f; }
    else if (o == 13) { w1 = 1.0f; }
    else if (o == 14) { w2 = 1.0f; }
}

// One 32-block pair: 2 groups of 16 blocks, 4 WMMAs, per-half merge, b96 store.
__device__ __forceinline__ void process_pair(
    const float* __restrict__ bp, float* __restrict__ op,
    int ldOfs, int stOfs, bool hi, v2f A1, v2f A2)
{
    v2fa t0 = *(const v2fa*)(bp + ldOfs);        // group0: lo=(b0,b1) hi=(b2,next)
    v2fa t1 = *(const v2fa*)(bp + 48 + ldOfs);   // group1
    v2f B0; B0.x = t0.x; B0.y = hi ? 1.0f : t0.y;   // K=3 row = 1.0 (bias trick)
    v2f B1; B1.x = t1.x; B1.y = hi ? 1.0f : t1.y;

    v8f c = {};
    v8f d1a = __builtin_amdgcn_wmma_f32_16x16x4_f32(false, A1, false, B0, (short)0, c, false, false);
    v8f d2a = __builtin_amdgcn_wmma_f32_16x16x4_f32(false, A2, false, B0, (short)0, c, false, false);
    v8f d1b = __builtin_amdgcn_wmma_f32_16x16x4_f32(false, A1, false, B1, (short)0, c, false, false);
    v8f d2b = __builtin_amdgcn_wmma_f32_16x16x4_f32(false, A2, false, B1, (short)0, c, false, false);

    // Lane n owns g0-block n, lane n+16 owns g1-block n: per-half select, no shuffles.
    float o[15];
    #pragma unroll
    for (int r = 0; r < 8; ++r) o[r]     = hi ? d1b[r] : d1a[r];   // outputs 0-7
    #pragma unroll
    for (int r = 0; r < 7; ++r) o[8 + r] = hi ? d2b[r] : d2a[r];   // outputs 8-14

    // o[0..2]=syndromes, o[3+3j+k]=corr_out[j][k], o[12..14]=raw block
    const float s0 = o[0], s1 = o[1], s2 = o[2];
    const float n2    = fmaf(s0, s0, fmaf(s1, s1, s2 * s2));
    const float scale = (n2 > ERR_T2) ? -CORR_S : 0.0f;
    const float c0 = fmaf(s0, o[3], fmaf(s1, o[6], s2 * o[9]));
    const float c1 = fmaf(s0, o[4], fmaf(s1, o[7], s2 * o[10]));
    const float c2 = fmaf(s0, o[5], fmaf(s1, o[8], s2 * o[11]));

    v2fa rr; rr.x = fmaf(scale, c0, o[12]); rr.y = fmaf(scale, c1, o[13]);
    *(v2fa*)(op + stOfs) = rr;                 // merges with next store -> b96
    op[stOfs + 2] = fmaf(scale, c2, o[14]);
}

__global__ __launch_bounds__(256) void qec_wmma_main(
    const float* __restrict__ params,
    const float* __restrict__ Ws, const float* __restrict__ bs,
    const float* __restrict__ Wc, const float* __restrict__ bc,
    float* __restrict__ out, long long fastUnits)   // unit = 64 blocks = 192 floats
{
    const int  lane = threadIdx.x & 31;
    const bool hi   = lane >= 16;           // supplies K=2,3 of B / sees D rows 8-15
    const int  key  = lane & 7;

    // A1 rows m and m+8 both hold output 'key'   -> every lane sees outputs 0-7.
    // A2 rows m and m+8 both hold output '8+key' -> every lane sees outputs 8-14 (15=zero).
    float w0, w1, w2, bb;
    out_row(key, Ws, bs, Wc, bc, w0, w1, w2, bb);
    v2f A1; A1.x = hi ? w2 : w0; A1.y = hi ? bb : w1;
    out_row(8 + key, Ws, bs, Wc, bc, w0, w1, w2, bb);
    v2f A2; A2.x = hi ? w2 : w0; A2.y = hi ? bb : w1;

    const int wavesPerBlk = blockDim.x >> 5;
    int wid = blockIdx.x * wavesPerBlk + (threadIdx.x >> 5);
    wid = __builtin_amdgcn_readfirstlane(wid);                 // force SGPR addressing
    const long long wavesTotal = (long long)gridDim.x * wavesPerBlk;

    const int ldOfs = (lane & 15) * 3 + (hi ? 2 : 0);  // loop-invariant lane offsets
    const int stOfs = lane * 3;

    const float* bp = params + (long long)wid * 192;
    float*       op = out    + (long long)wid * 192;
    const long long strideF = wavesTotal * 192;

    for (long long u = wid; u < fastUnits; u += wavesTotal, bp += strideF, op += strideF) {
        if (u + wavesTotal < fastUnits)
            __builtin_prefetch(bp + strideF + lane * 6, 0, 0);   // next unit, every line

        process_pair(bp,      op,      ldOfs, stOfs, hi, A1, A2);   // blocks  0..31
        process_pair(bp + 96, op + 96, ldOfs, stOfs, hi, A1, A2);   // blocks 32..63
    }
}

// Scalar tail: remaining <=127 blocks (incl. ragged last block, zero-padded like reference).
__global__ __launch_bounds__(128) void qec_tail(
    const float* __restrict__ params,
    const float* __restrict__ Ws, const float* __restrict__ bs,
    const float* __restrict__ Wc, const float* __restrict__ bc,
    float* __restrict__ out, long long nElem, long long startBlk, long long nBlocks)
{
    long long b = startBlk + (long long)blockIdx.x * blockDim.x + threadIdx.x;
    if (b >= nBlocks) return;
    const long long e = b * 3;
    const float x0 = (e     < nElem) ? params[e]     : 0.0f;
    const float x1 = (e + 1 < nElem) ? params[e + 1] : 0.0f;
    const float x2 = (e + 2 < nElem) ? params[e + 2] : 0.0f;

    float syn[3];
    #pragma unroll
    for (int k = 0; k < 3; ++k)
        syn[k] = fmaf(Ws[k*3], x0, fmaf(Ws[k*3+1], x1, fmaf(Ws[k*3+2], x2, bs[k])));
    const float n2    = fmaf(syn[0], syn[0], fmaf(syn[1], syn[1], syn[2]*syn[2]));
    const float scale = (n2 > ERR_T2) ? -CORR_S : 0.0f;

    float corr[3] = {0.0f, 0.0f, 0.0f};
    #pragma unroll
    for (int j = 0; j < 3; ++j)
        #pragma unroll
        for (int k = 0; k < 3; ++k) {
            const int q = j * 3 + k;
            const float co = fmaf(Wc[q*3], x0, fmaf(Wc[q*3+1], x1, fmaf(Wc[q*3+2], x2, bc[q])));
            corr[k] = fmaf(syn[j], co, corr[k]);
        }
    if (e     < nElem) out[e]     = fmaf(scale, corr[0], x0);
    if (e + 1 < nElem) out[e + 1] = fmaf(scale, corr[1], x1);
    if (e + 2 < nElem) out[e + 2] = fmaf(scale, corr[2], x2);
}

extern "C" void kernel_launch(void* const* d_in, const int* in_sizes, int n_in,
                              void* d_out, int out_size, void* d_ws, size_t ws_size,
                              hipStream_t stream) {
    const float* params = (const float*)d_in[0];
    const float* Ws     = (const float*)d_in[1];
    const float* bs     = (const float*)d_in[2];
    const float* Wc     = (const float*)d_in[3];
    const float* bc     = (const float*)d_in[4];
    float* out = (float*)d_out;

    const long long n       = (long long)in_sizes[0];
    const long long nBlocks = (n + 2) / 3;                 // ceil: reference zero-pads

    // Unit u (64 blocks, 192 floats) is guard-free iff u*192 + 193 <= n
    // (hi-lane b64 reads reach 1 float past the unit).
    long long fastUnits = (n >= 193) ? ((n - 193) / 192 + 1) : 0;
    if (fastUnits * 64 > nBlocks) fastUnits = nBlocks / 64;  // safety

    if (fastUnits > 0) {
        const int threads = 256, wavesPerBlk = 8;
        long long blocksNeeded = (fastUnits + wavesPerBlk - 1) / wavesPerBlk;
        int grid = (int)(blocksNeeded < 16384 ? blocksNeeded : 16384);
        qec_wmma_main<<<grid, threads, 0, stream>>>(params, Ws, bs, Wc, bc, out, fastUnits);
    }

    const long long startBlk   = fastUnits * 64;
    const long long tailBlocks = nBlocks - startBlk;
    if (tailBlocks > 0) {
        const int tthreads = 128;
        int tgrid = (int)((tailBlocks + tthreads - 1) / tthreads);
        qec_tail<<<tgrid, tthreads, 0, stream>>>(params, Ws, bs, Wc, bc, out,
                                                 n, startBlk, nBlocks);
    }
}